// MGD_loss_eye_85323820302377
// MI455X (gfx1250) — compile-verified
//
#include <hip/hip_runtime.h>
#include <math.h>

typedef __attribute__((ext_vector_type(2))) float v2f;
typedef __attribute__((ext_vector_type(4))) float v4f;
typedef __attribute__((ext_vector_type(8))) float v8f;

#define RED_BLOCKS  2048
#define RED_THREADS 256
#define WAVES_PER_BLOCK (RED_THREADS / 32)

// Cross-lane wave32 reduction using the CDNA5 f32 WMMA.
// A (16x4): lanes 0-15 supply A[m][0] (VGPR0) / A[m][1] (VGPR1),
//           lanes 16-31 supply A[m][2] / A[m][3].
// With a[0]=partial, a[1]=0 and B = all-ones (4x16),
// D[m][n] = partial[m] + partial[m+16]  (independent of n).
// C/D layout: lane<16 holds M=0..7 in c[0..7]; lane>=16 holds M=8..15.
// So sum(c[0..7]) gives half-sums; one shfl_xor(16) completes the total.
__device__ __forceinline__ float wave_reduce_wmma(float partial) {
    v2f a; a.x = partial; a.y = 0.0f;
    v2f b; b.x = 1.0f;    b.y = 1.0f;
    v8f c = {};
    c = __builtin_amdgcn_wmma_f32_16x16x4_f32(
            /*neg_a=*/false, a, /*neg_b=*/false, b,
            /*c_mod=*/(short)0, c, /*reuse_a=*/false, /*reuse_b=*/false);
    float local = ((c[0] + c[1]) + (c[2] + c[3])) + ((c[4] + c[5]) + (c[6] + c[7]));
    float other = __shfl_xor(local, 16, 32);
    return local + other;  // full wave sum, all lanes
}

__device__ __forceinline__ float masked_sq_acc(v4f e, v4f yv, float acc) {
    float x0 = (yv.x != 0.0f) ? e.x : 0.0f;
    float x1 = (yv.y != 0.0f) ? e.y : 0.0f;
    float x2 = (yv.z != 0.0f) ? e.z : 0.0f;
    float x3 = (yv.w != 0.0f) ? e.w : 0.0f;
    acc = fmaf(x0, x0, acc);
    acc = fmaf(x1, x1, acc);
    acc = fmaf(x2, x2, acc);
    acc = fmaf(x3, x3, acc);
    return acc;
}

__global__ void __launch_bounds__(RED_THREADS)
mgd_pass1(const float* __restrict__ eps, const float* __restrict__ y,
          float* __restrict__ partials, long long n4, long long n) {
    __shared__ float wave_sums[WAVES_PER_BLOCK];

    const v4f* __restrict__ e4 = reinterpret_cast<const v4f*>(eps);
    const v4f* __restrict__ y4 = reinterpret_cast<const v4f*>(y);

    const long long tid    = (long long)blockIdx.x * blockDim.x + threadIdx.x;
    const long long stride = (long long)gridDim.x * blockDim.x;

    float acc0 = 0.0f;
    float acc1 = 0.0f;

    long long i = tid;
    // 2x unrolled: 4 independent nontemporal b128 loads in flight per iteration
    for (; i + stride < n4; i += 2 * stride) {
        v4f e0 = __builtin_nontemporal_load(&e4[i]);
        v4f v0 = __builtin_nontemporal_load(&y4[i]);
        v4f e1 = __builtin_nontemporal_load(&e4[i + stride]);
        v4f v1 = __builtin_nontemporal_load(&y4[i + stride]);
        acc0 = masked_sq_acc(e0, v0, acc0);
        acc1 = masked_sq_acc(e1, v1, acc1);
    }
    if (i < n4) {
        v4f e0 = __builtin_nontemporal_load(&e4[i]);
        v4f v0 = __builtin_nontemporal_load(&y4[i]);
        acc0 = masked_sq_acc(e0, v0, acc0);
    }
    float acc = acc0 + acc1;

    // scalar tail (n not divisible by 4) -- handled by a single thread
    if (blockIdx.x == 0 && threadIdx.x == 0) {
        for (long long t = n4 * 4; t < n; ++t) {
            float x = (y[t] != 0.0f) ? eps[t] : 0.0f;
            acc = fmaf(x, x, acc);
        }
    }

    float wsum = wave_reduce_wmma(acc);
    const int wave = threadIdx.x >> 5;
    const int lane = threadIdx.x & 31;
    if (lane == 0) wave_sums[wave] = wsum;
    __syncthreads();
    if (threadIdx.x == 0) {
        float s = 0.0f;
        #pragma unroll
        for (int w = 0; w < WAVES_PER_BLOCK; ++w) s += wave_sums[w];
        partials[blockIdx.x] = s;
    }
}

__global__ void __launch_bounds__(RED_THREADS)
mgd_pass2(const float* __restrict__ partials, const float* __restrict__ sigma,
          float* __restrict__ out, int nparts, float Bf, float Df) {
    __shared__ float wave_sums[WAVES_PER_BLOCK];

    float acc = 0.0f;
    for (int i = threadIdx.x; i < nparts; i += blockDim.x) acc += partials[i];

    float wsum = wave_reduce_wmma(acc);
    const int wave = threadIdx.x >> 5;
    const int lane = threadIdx.x & 31;
    if (lane == 0) wave_sums[wave] = wsum;
    __syncthreads();
    if (threadIdx.x == 0) {
        float total = 0.0f;
        #pragma unroll
        for (int w = 0; w < WAVES_PER_BLOCK; ++w) total += wave_sums[w];

        // softplus(sigma), numerically stable, f32 like the reference
        float sg = sigma[0];
        float sp = fmaxf(sg, 0.0f) + log1pf(expf(-fabsf(sg)));

        const float LOG_2PI = 1.8378770664093454836f; // ln(2*pi)
        out[0] = 0.5f * (total / (Bf * sp) + Df * (LOG_2PI + logf(sp)));
    }
}

extern "C" void kernel_launch(void* const* d_in, const int* in_sizes, int n_in,
                              void* d_out, int out_size, void* d_ws, size_t ws_size,
                              hipStream_t stream) {
    const float* eps   = (const float*)d_in[0];
    const float* y     = (const float*)d_in[1];
    const float* sigma = (const float*)d_in[2];
    float* out      = (float*)d_out;
    float* partials = (float*)d_ws;

    const long long n  = (long long)in_sizes[0];   // B * Q * N
    const long long n4 = n / 4;

    int nblocks = RED_BLOCKS;
    long long ws_cap = (long long)(ws_size / sizeof(float));
    if (ws_cap < nblocks) nblocks = (ws_cap > 0) ? (int)ws_cap : 1;

    const float Bf = 256.0f;               // BATCH from the reference
    const float Df = (float)(n / 256);     // event dim D = Q*N

    mgd_pass1<<<nblocks, RED_THREADS, 0, stream>>>(eps, y, partials, n4, n);
    mgd_pass2<<<1, RED_THREADS, 0, stream>>>(partials, sigma, out, nblocks, Bf, Df);
}